// CenterLoss_22136261443658
// MI455X (gfx1250) — compile-verified
//
#include <hip/hip_runtime.h>
#include <hip/hip_bf16.h>

#define N_SAMPLES 16384
#define N_CLASSES 10000
#define DIM       128
#define LRATE     0.5f

typedef __attribute__((ext_vector_type(2))) float v2f;
typedef __attribute__((ext_vector_type(8))) float v8f;

// ---------------- Kernel 0: zero the accumulators (S[C,D] and cnt[C]) ----------
__global__ void cl_zero(float* __restrict__ buf, int n) {
    int i = blockIdx.x * blockDim.x + threadIdx.x;
    if (i < n) buf[i] = 0.0f;
}

// ---------------- Kernel 1: stream 655MB one-hot labels -> idx[n] --------------
// Pure bandwidth: float4 grid-stride scan; nonzero hit is 1-in-10000 so the
// branch body almost never executes. Prefetch next stride chunk (global_prefetch).
__global__ void cl_extract_idx(const float* __restrict__ labels,
                               int* __restrict__ idxbuf) {
    const long long totalq = (long long)N_SAMPLES * N_CLASSES / 4;
    long long q      = (long long)blockIdx.x * blockDim.x + threadIdx.x;
    long long stride = (long long)gridDim.x * blockDim.x;
    const float4* lp = (const float4*)labels;
    for (; q < totalq; q += stride) {
        __builtin_prefetch((const void*)(lp + q + stride), 0, 0);
        float4 f = lp[q];
        if ((f.x != 0.f) | (f.y != 0.f) | (f.z != 0.f) | (f.w != 0.f)) {
            long long base = q * 4;                 // C % 4 == 0: no row straddle
            int row = (int)(base / N_CLASSES);
            int col = (int)(base % N_CLASSES);
            if (f.x != 0.f) idxbuf[row] = col;
            if (f.y != 0.f) idxbuf[row] = col + 1;
            if (f.z != 0.f) idxbuf[row] = col + 2;
            if (f.w != 0.f) idxbuf[row] = col + 3;
        }
    }
}

// ---------------- Kernel 2: WMMA tile-compacted scatter of labels.T @ preds ----
// One wave32 per 16-sample tile. Build selection matrix A (16 class-slots x 16
// samples, first-occurrence compaction), P = A @ preds_tile via
// V_WMMA_F32_16X16X4_F32 chained over K=16, then one atomic row per distinct
// class. Also accumulates per-class counts.
__global__ void cl_scatter_wmma(const float* __restrict__ preds,
                                const int*   __restrict__ idxbuf,
                                float* __restrict__ S,
                                float* __restrict__ cnt) {
    const int lane = threadIdx.x & 31;
    const int wave = threadIdx.x >> 5;
    const int tile = blockIdx.x * (blockDim.x >> 5) + wave;
    const int s0   = tile * 16;
    const int m    = lane & 15;

    const int myidx = idxbuf[s0 + m];          // lanes 16..31 mirror lanes 0..15
    if (lane < 16) atomicAdd(&cnt[myidx], 1.0f);

    // first-occurrence flag for this tile-slot
    int first = 1;
    #pragma unroll 1
    for (int j = 0; j < m; ++j) {
        if (__shfl(myidx, j, 32) == myidx) first = 0;
    }

    // A fragments: 16x4 layout — lane<16 holds K = k0+0,k0+1; lane>=16 holds k0+2,k0+3
    const int chalf = (lane >> 4) << 1;
    v2f afrag[4];
    #pragma unroll
    for (int k = 0; k < 4; ++k) {
        const int c0 = 4 * k + chalf;
        const int i0 = __shfl(myidx, c0,     32);
        const int i1 = __shfl(myidx, c0 + 1, 32);
        afrag[k].x = (first && i0 == myidx) ? 1.0f : 0.0f;
        afrag[k].y = (first && i1 == myidx) ? 1.0f : 0.0f;
    }

    const int dlane = lane & 15;
    const int mbase = (lane < 16) ? 0 : 8;

    // Hoist dc-invariant shuffles + atomic row base addresses out of the hot loop.
    // D layout: VGPR v -> row M = mbase+v, column dim = dc*16 + dlane.
    int    fov[8];
    float* srow[8];
    #pragma unroll
    for (int v = 0; v < 8; ++v) {
        const int M   = mbase + v;
        const int cls = __shfl(myidx, M, 32);
        fov[v]  = __shfl(first, M, 32);
        srow[v] = S + (size_t)cls * DIM + dlane;   // + dc*16 folds into offset:
    }

    #pragma unroll 2
    for (int dc = 0; dc < 8; ++dc) {
        const int dim = dc * 16 + dlane;
        v8f acc = {};
        #pragma unroll
        for (int k = 0; k < 4; ++k) {
            const int c0 = 4 * k + chalf;      // B 4x16 layout mirrors A halves
            v2f b;
            b.x = preds[(size_t)(s0 + c0)     * DIM + dim];
            b.y = preds[(size_t)(s0 + c0 + 1) * DIM + dim];
            acc = __builtin_amdgcn_wmma_f32_16x16x4_f32(
                      false, afrag[k], false, b, (short)0, acc, false, false);
        }
        union { v8f v; float f[8]; } u; u.v = acc;
        #pragma unroll
        for (int v = 0; v < 8; ++v) {
            if (fov[v]) atomicAdd(srow[v] + dc * 16, u.f[v]);
        }
    }
}

// ---------------- Kernel 3: updated = center - LR*(cnt*center - S)/(cnt+1) -----
__global__ void cl_update(const float* __restrict__ center,
                          const float* __restrict__ S,
                          const float* __restrict__ cnt,
                          float* __restrict__ out) {
    const int total = N_CLASSES * DIM / 4;
    int i = blockIdx.x * blockDim.x + threadIdx.x;     // float4 index
    if (i >= total) return;
    const int c   = i / (DIM / 4);
    const float n = cnt[c];
    const float w = LRATE / (n + 1.0f);
    float4 ce = ((const float4*)center)[i];
    float4 s  = ((const float4*)S)[i];
    float4 r;
    r.x = ce.x - w * (n * ce.x - s.x);
    r.y = ce.y - w * (n * ce.y - s.y);
    r.z = ce.z - w * (n * ce.z - s.z);
    r.w = ce.w - w * (n * ce.w - s.w);
    ((float4*)out)[i] = r;
}

extern "C" void kernel_launch(void* const* d_in, const int* in_sizes, int n_in,
                              void* d_out, int out_size, void* d_ws, size_t ws_size,
                              hipStream_t stream) {
    const float* preds  = (const float*)d_in[0];   // [N, D] f32
    const float* labels = (const float*)d_in[1];   // [N, C] f32 one-hot (dense)
    const float* center = (const float*)d_in[2];   // [C, D] f32
    float* out = (float*)d_out;                    // [C, D] f32

    // workspace layout
    int*   idxbuf = (int*)d_ws;                                   // 64 KB
    float* cnt    = (float*)((char*)d_ws + (64 << 10));           // 40 KB (pad to 64)
    float* S      = (float*)((char*)d_ws + (128 << 10));          // 5.12 MB

    // 0) zero cnt + S
    cl_zero<<<(N_CLASSES + 255) / 256, 256, 0, stream>>>(cnt, N_CLASSES);
    cl_zero<<<(N_CLASSES * DIM + 255) / 256, 256, 0, stream>>>(S, N_CLASSES * DIM);

    // 1) extract indices from dense one-hot (bandwidth-bound: ~655 MB)
    cl_extract_idx<<<8192, 256, 0, stream>>>(labels, idxbuf);

    // 2) WMMA tile-compacted scatter: S[c] += sum preds, cnt[c] += 1
    {
        const int tiles = N_SAMPLES / 16;          // 1024 tiles, 8 waves/block
        cl_scatter_wmma<<<tiles / 8, 256, 0, stream>>>(preds, idxbuf, S, cnt);
    }

    // 3) fused update
    {
        const int total4 = N_CLASSES * DIM / 4;
        cl_update<<<(total4 + 255) / 256, 256, 0, stream>>>(center, S, cnt, out);
    }
}